// BatchInvariantLinear_33303176413804
// MI455X (gfx1250) — compile-verified
//
#include <hip/hip_runtime.h>
#include <hip/hip_bf16.h>

// CDNA5 / gfx1250: wave32, WMMA (not MFMA), async global->LDS copies.

typedef __attribute__((ext_vector_type(2))) float v2f;
typedef __attribute__((ext_vector_type(8))) float v8f;

#define KT 32                 // K slice per LDS stage
#define LDA 36                // padded row stride in floats (144B: 16B-aligned, bank-conflict-free)
#define TILE_FLOATS (128 * LDA)

// Async copy 16 bytes/lane from global (SGPR base + per-lane byte offset, GVS mode)
// into LDS at per-lane byte address. Tracked by ASYNCcnt.
__device__ __forceinline__ void async_b128(unsigned lds_byte, unsigned glob_byte_off,
                                           unsigned long long base) {
    asm volatile("global_load_async_to_lds_b128 %0, %1, %2"
                 :
                 : "v"(lds_byte), "v"(glob_byte_off), "s"(base)
                 : "memory");
}

__device__ __forceinline__ void wait_async0() {
    asm volatile("s_wait_asynccnt 0x0" ::: "memory");
}

__global__ __launch_bounds__(256)
void BatchInvariantLinear_gemm_f32_wmma(const float* __restrict__ x,
                                        const float* __restrict__ w,
                                        const float* __restrict__ bias,
                                        float* __restrict__ out,
                                        int M, int N, int K) {
    extern __shared__ float smem[];
    // Low 32 bits of a flat LDS pointer == LDS byte address (LDS aperture, ISA 10.2).
    const unsigned lds_base = (unsigned)(unsigned long long)(void*)smem;

    const int tid   = threadIdx.x;
    const int lane  = tid & 31;
    const int wave  = tid >> 5;           // 0..7
    const int mBase = blockIdx.y * 128;
    const int nBase = blockIdx.x * 128;

    const int lm    = lane & 15;          // row-in-subtile for fragments
    const int khalf = (lane >> 4) << 1;   // 0 or 2: K pair held by this half-wave
    const int aRow  = (wave >> 2) * 64;   // wave M offset inside WG tile (0 or 64)
    const int bRow  = (wave & 3) * 32;    // wave N offset inside WG tile (0..96)

    v8f acc[4][2];
#pragma unroll
    for (int m0 = 0; m0 < 4; ++m0)
#pragma unroll
        for (int n0 = 0; n0 < 2; ++n0)
            acc[m0][n0] = v8f{0.f, 0.f, 0.f, 0.f, 0.f, 0.f, 0.f, 0.f};

    const unsigned long long xb = (unsigned long long)x;
    const unsigned long long wb = (unsigned long long)w;

    // Issue one stage of async copies: A = x[mBase..+127, k0..+31], B = w[nBase..+127, k0..+31].
    // Per wave: 16 rows of A and 16 rows of B, 8 lanes x b128 per row -> 8 async ops/wave.
    auto load_stage = [&](int kt, int s) {
        const unsigned aoff = lds_base + (unsigned)(s * 2 * TILE_FLOATS) * 4u;
        const unsigned boff = aoff + (unsigned)TILE_FLOATS * 4u;
        const int k0   = kt * KT;
        const int colf = (lane & 7) * 4;  // float column within the 32-wide slice
        const int rsub = lane >> 3;       // 0..3
#pragma unroll
        for (int j = 0; j < 4; ++j) {
            const int row = wave * 16 + j * 4 + rsub;          // 0..127
            const unsigned lrow = (unsigned)(row * LDA + colf) * 4u;
            const unsigned ga = (unsigned)(((size_t)(mBase + row) * K + k0 + colf) * 4);
            async_b128(aoff + lrow, ga, xb);
            const unsigned gb = (unsigned)(((size_t)(nBase + row) * K + k0 + colf) * 4);
            async_b128(boff + lrow, gb, wb);
        }
    };

    const int ktiles = K / KT;

    load_stage(0, 0);
    wait_async0();
    __syncthreads();

    for (int kt = 0; kt < ktiles; ++kt) {
        const int cur = kt & 1;
        if (kt + 1 < ktiles) load_stage(kt + 1, cur ^ 1);  // prefetch next slice (async)

        const float* As = smem + cur * 2 * TILE_FLOATS;
        const float* Bs = As + TILE_FLOATS;

#pragma unroll
        for (int kk = 0; kk < KT; kk += 4) {
            const int kc = kk + khalf;
            v2f af[4], bf[2];
#pragma unroll
            for (int m0 = 0; m0 < 4; ++m0)
                af[m0] = *(const v2f*)(As + (aRow + m0 * 16 + lm) * LDA + kc);
#pragma unroll
            for (int n0 = 0; n0 < 2; ++n0)
                bf[n0] = *(const v2f*)(Bs + (bRow + n0 * 16 + lm) * LDA + kc);
#pragma unroll
            for (int m0 = 0; m0 < 4; ++m0)
#pragma unroll
                for (int n0 = 0; n0 < 2; ++n0)
                    acc[m0][n0] = __builtin_amdgcn_wmma_f32_16x16x4_f32(
                        false, af[m0], false, bf[n0],
                        (short)0, acc[m0][n0], false, false);
        }

        wait_async0();     // my prefetch for next stage has landed in LDS
        __syncthreads();   // everyone's prefetch visible; safe to switch buffers
    }

    // Epilogue: D 16x16 f32 layout: lanes 0-15 -> M rows 0..7 (vgpr i), lanes 16-31 -> rows 8..15.
#pragma unroll
    for (int m0 = 0; m0 < 4; ++m0) {
        const int rbase = mBase + aRow + m0 * 16 + ((lane >> 4) << 3);
#pragma unroll
        for (int n0 = 0; n0 < 2; ++n0) {
            const int col = nBase + bRow + n0 * 16 + lm;
            const float bv = bias[col];
#pragma unroll
            for (int i = 0; i < 8; ++i)
                out[(size_t)(rbase + i) * N + col] = acc[m0][n0][i] + bv;
        }
    }
}

extern "C" void kernel_launch(void* const* d_in, const int* in_sizes, int n_in,
                              void* d_out, int out_size, void* d_ws, size_t ws_size,
                              hipStream_t stream) {
    const float* x    = (const float*)d_in[0];
    const float* w    = (const float*)d_in[1];
    const float* bias = (const float*)d_in[2];
    float* out        = (float*)d_out;

    const int OUT = in_sizes[2];           // 4096
    const int IN  = in_sizes[1] / OUT;     // 4096
    const int M   = in_sizes[0] / IN;      // B*S = 8192

    dim3 grid((unsigned)(OUT / 128), (unsigned)(M / 128));
    dim3 block(256);
    const size_t shmem = 2u * 2u * TILE_FLOATS * sizeof(float);  // double-buffered A+B = 72 KB

    BatchInvariantLinear_gemm_f32_wmma<<<grid, block, shmem, stream>>>(
        x, w, bias, out, M, OUT, IN);
}